// GCNext_7825430413347
// MI455X (gfx1250) — compile-verified
//
#include <hip/hip_runtime.h>

// GCNext fused scan for MI455X (gfx1250, wave32).
// One workgroup per batch element; h[b] lives in LDS across all 48 layers.
// Skeletal 22x22 conv runs on V_WMMA_F32_16X16X4_F32 (fp32-exact tensor op).
// Tridiagonal temporal weights pre-compacted into d_ws to avoid strided gathers.
// D-tile scatter is branchless: invalid lanes RMW dump slots via address select.

#define NJ    22
#define NV    66
#define NT    50
#define NL    48
#define NB    2048
#define HSTR  64      // padded row stride of hs (t dimension padded 50->64)
#define HROWS 72      // padded v rows (need up to v=71 for K-padding reads)

typedef float v2f __attribute__((ext_vector_type(2)));
typedef float v8f __attribute__((ext_vector_type(8)));

// H36M skeleton adjacency (identity + symmetrized bones), row bitmasks.
__constant__ unsigned int SKL_BITS[NJ] = {
    0x000103u, 0x000007u, 0x00000Eu, 0x00000Cu, 0x000130u, 0x000070u,
    0x0000E0u, 0x0000C0u, 0x000311u, 0x021700u, 0x000E00u, 0x000C00u,
    0x003200u, 0x007000u, 0x01E000u, 0x00C000u, 0x014000u, 0x060200u,
    0x0E0000u, 0x3C0000u, 0x180000u, 0x280000u
};

// ---- prep: compact adj_tj tridiagonals: out[l][v][k][t], k=0 sub, k=1 sup ----
__global__ void prep_atj(const float* __restrict__ atj, float* __restrict__ out) {
    int i = blockIdx.x * 256 + threadIdx.x;           // NL*NV*2*NT
    if (i >= NL * NV * 2 * NT) return;
    int t = i % NT;
    int k = (i / NT) & 1;
    int v = (i / (2 * NT)) % NV;
    int l = i / (2 * NT * NV);
    int src = t + (k ? 1 : -1);
    float val = 0.f;
    if (src >= 0 && src < NT)
        val = atj[(((size_t)l * NV + v) * NT + t) * NT + src];
    out[i] = val;
}

// ---- prep: compact temp_w tridiagonals: out[l][k][t] ----
__global__ void prep_tw(const float* __restrict__ tw, float* __restrict__ out) {
    int i = blockIdx.x * 256 + threadIdx.x;           // NL*2*NT
    if (i >= NL * 2 * NT) return;
    int t = i % NT;
    int k = (i / NT) & 1;
    int l = i / (2 * NT);
    int src = t + (k ? 1 : -1);
    out[i] = (src >= 0 && src < NT) ? tw[((size_t)l * NT + t) * NT + src] : 0.f;
}

// ---- prep: mask + zero-pad skeleton weights to [l][32][24] for WMMA tiles ----
__global__ void prep_wsk(const float* __restrict__ skl, float* __restrict__ out) {
    int i = blockIdx.x * 256 + threadIdx.x;           // NL*32*24
    if (i >= NL * 32 * 24) return;
    int c = i % 24;
    int r = (i / 24) % 32;
    int l = i / (24 * 32);
    float val = 0.f;
    if (r < NJ && c < NJ && ((SKL_BITS[r] >> c) & 1u))
        val = skl[((size_t)l * NJ + r) * NJ + c];
    out[i] = val;
}

// ---- main fused kernel: one block per batch element, full 48-layer scan ----
__global__ __launch_bounds__(256) void gcnext_fused(
    const float* __restrict__ x,     const float* __restrict__ adj_jc,
    const float* __restrict__ alpha, const float* __restrict__ beta,
    const float* __restrict__ atjc,  const float* __restrict__ twc,
    const float* __restrict__ wskc,  float* __restrict__ out)
{
    __shared__ float hs[HROWS * HSTR];       // h tile, zero-padded rows/cols
    __shared__ float ys[NV * NT + 8];        // pre-norm output + 8 dump slots
    __shared__ float atjl[NV * 2 * NT];      // per-vertex tridiagonal weights
    __shared__ float wskl[32 * 24];          // padded masked skeleton weights
    __shared__ float ajcl[NJ * 9];           // 3x3 coordinate mixers
    __shared__ float twl[2 * NT];            // shared tridiagonal weights
    __shared__ float al[NV], bl[NV];         // LN affine
    __shared__ float mn[NT], rstd[NT];       // LN stats per t

    const int tid  = threadIdx.x;
    const int lane = tid & 31;
    const int wv   = tid >> 5;
    const size_t xoff = (size_t)blockIdx.x * (NV * NT);

    for (int i = tid; i < HROWS * HSTR; i += 256) hs[i] = 0.f;
    __syncthreads();
    for (int i = tid; i < NV * NT; i += 256) {
        int v = i / NT, t = i - v * NT;
        hs[v * HSTR + t] = x[xoff + i];
    }
    __syncthreads();

    for (int l = 0; l < NL; ++l) {
        // ---- stage this layer's weights into LDS (L2-resident, coalesced) ----
        for (int i = tid; i < NV * 2 * NT; i += 256) {
            atjl[i] = atjc[(size_t)l * (NV * 2 * NT) + i];
            if (l + 1 < NL)   // warm L2 for next layer's big weight block
                __builtin_prefetch(&atjc[(size_t)(l + 1) * (NV * 2 * NT) + i], 0, 1);
        }
        for (int i = tid; i < 32 * 24; i += 256) wskl[i] = wskc[l * 32 * 24 + i];
        if (tid < NJ * 9) ajcl[tid] = adj_jc[l * (NJ * 9) + tid];
        if (tid < 2 * NT) twl[tid]  = twc[l * 2 * NT + tid];
        if (tid < NV) { al[tid] = alpha[l * NV + tid]; bl[tid] = beta[l * NV + tid]; }
        __syncthreads();

        // ---- phase 0: jc (3x3) + tc + tj (tridiagonal, sparse) -> ys ----
        for (int i = tid; i < NV * NT; i += 256) {
            int v = i / NT, t = i - v * NT;
            int j = v / 3, k = v - j * 3;
            const float* aj = &ajcl[j * 9 + k * 3];
            const float* hj = &hs[(j * 3) * HSTR + t];
            float acc = aj[0] * hj[0] + aj[1] * hj[HSTR] + aj[2] * hj[2 * HSTR];
            int tm = (t == 0)      ? 0      : t - 1;   // weight is 0 at the clamp
            int tp = (t == NT - 1) ? NT - 1 : t + 1;
            float hm = hs[v * HSTR + tm];
            float hp = hs[v * HSTR + tp];
            acc += twl[t] * hm + twl[NT + t] * hp;
            acc += atjl[v * 2 * NT + t] * hm + atjl[v * 2 * NT + NT + t] * hp;
            ys[i] = acc;
        }
        __syncthreads();

        // ---- phase 1: skeletal conv via V_WMMA_F32_16X16X4_F32 ----
        // 24 output tiles (c:3 x mt:2 x nt:4), 3 tiles per wave; uniform wave
        // control flow so EXEC is all-1s at every WMMA.
        {
            const int half2 = (lane < 16) ? 0 : 2;   // A/B K-half per ISA layout
            const int ml    = lane & 15;
            for (int tl = wv * 3; tl < wv * 3 + 3; ++tl) {
                int c  = tl >> 3;
                int mt = (tl & 7) >> 2;
                int nt = tl & 3;
                int mrow = mt * 16 + ml;             // A row (padded to 32)
                int col  = nt * 16 + ml;             // B/D column (padded to 64)
                v8f acc = {0.f, 0.f, 0.f, 0.f, 0.f, 0.f, 0.f, 0.f};
#pragma unroll
                for (int k0 = 0; k0 < 24; k0 += 4) {
                    int ka = k0 + half2;
                    v2f a, bfr;
                    a.x = wskl[mrow * 24 + ka];          // A: VGPR0 K=ka, VGPR1 K=ka+1
                    a.y = wskl[mrow * 24 + ka + 1];
                    int vb = ka * 3 + c;                 // B rows = joints at coord c
                    bfr.x = hs[vb * HSTR + col];         // B: VGPR0 rows k0/k0+2
                    bfr.y = hs[(vb + 3) * HSTR + col];   //    VGPR1 rows k0+1/k0+3
                    acc = __builtin_amdgcn_wmma_f32_16x16x4_f32(
                        false, a, false, bfr, (short)0, acc, false, false);
                }
                // D layout: VGPR r -> M = r (lanes 0-15) / r+8 (lanes 16-31).
                // Branchless scatter: invalid lanes/rows RMW dump slot instead.
                int mbase   = mt * 16 + ((lane >> 4) << 3);
                int colOk   = (col < NT);
#pragma unroll
                for (int r = 0; r < 8; ++r) {
                    int mr    = mbase + r;
                    int valid = colOk & (mr < NJ);
                    int addr  = valid ? ((mr * 3 + c) * NT + col) : (NV * NT + r);
                    ys[addr] += acc[r];
                }
            }
        }
        __syncthreads();

        // ---- phase 2: LayerNorm over V per (b,t) ----
        if (tid < NT) {
            float s = 0.f, s2 = 0.f;
            for (int v = 0; v < NV; ++v) {
                float y = ys[v * NT + tid];
                s += y; s2 += y * y;
            }
            float m   = s * (1.f / NV);
            float var = s2 * (1.f / NV) - m * m;
            mn[tid]   = m;
            rstd[tid] = rsqrtf(var + 1e-5f);
        }
        __syncthreads();

        // ---- residual update ----
        for (int i = tid; i < NV * NT; i += 256) {
            int v = i / NT, t = i - v * NT;
            hs[v * HSTR + t] += (ys[i] - mn[t]) * rstd[t] * al[v] + bl[v];
        }
        __syncthreads();
    }

    for (int i = tid; i < NV * NT; i += 256) {
        int v = i / NT, t = i - v * NT;
        out[xoff + i] = hs[v * HSTR + t];
    }
}

extern "C" void kernel_launch(void* const* d_in, const int* in_sizes, int n_in,
                              void* d_out, int out_size, void* d_ws, size_t ws_size,
                              hipStream_t stream) {
    const float* x      = (const float*)d_in[0];
    const float* adj_jc = (const float*)d_in[1];
    const float* skl_w  = (const float*)d_in[2];
    const float* temp_w = (const float*)d_in[3];
    const float* adj_tj = (const float*)d_in[4];
    const float* alpha  = (const float*)d_in[5];
    const float* beta   = (const float*)d_in[6];

    float* ws   = (float*)d_ws;
    float* atjc = ws;                                  // NL*NV*2*NT = 316800 floats
    float* twc  = atjc + (size_t)NL * NV * 2 * NT;     // NL*2*NT   = 4800
    float* wskc = twc + NL * 2 * NT;                   // NL*32*24  = 36864

    prep_atj<<<(NL * NV * 2 * NT + 255) / 256, 256, 0, stream>>>(adj_tj, atjc);
    prep_tw <<<(NL * 2 * NT + 255) / 256,      256, 0, stream>>>(temp_w, twc);
    prep_wsk<<<(NL * 32 * 24 + 255) / 256,     256, 0, stream>>>(skl_w, wskc);

    gcnext_fused<<<NB, 256, 0, stream>>>(x, adj_jc, alpha, beta,
                                         atjc, twc, wskc, (float*)d_out);
}